// RelKDAdapter_60284160966709
// MI455X (gfx1250) — compile-verified
//
#include <hip/hip_runtime.h>
#include <hip/hip_bf16.h>

typedef __attribute__((ext_vector_type(2))) float v2f;
typedef __attribute__((ext_vector_type(8))) float v8f;

#define N_SRC 100000
#define N_DST 100000
#define NEDGE 640000
#define D_SRC 256
#define D_REL 128

// ---------------------------------------------------------------------------
// Zero-fill (d_out is poisoned with 0xAA; dst and deg sections are accumulated
// into with atomics, so they must start at 0.0f every call).
// ---------------------------------------------------------------------------
__global__ void zero_f32(float* __restrict__ p, int n) {
    int i = blockIdx.x * blockDim.x + threadIdx.x;
    if (i < n) p[i] = 0.0f;
}

// ---------------------------------------------------------------------------
// Degree count: one thread per edge. Sums of 1.0f are exact (< 2^24).
// ---------------------------------------------------------------------------
__global__ void count_deg(const int* __restrict__ row, float* __restrict__ deg, int e) {
    int i = blockIdx.x * blockDim.x + threadIdx.x;
    if (i < e) atomicAdd(&deg[row[i]], 1.0f);
}

// deg = max(deg, 1) — this is the final deg output; the scatter kernel reads
// it back to form 1/deg.
__global__ void clamp_deg(float* __restrict__ deg, int n) {
    int i = blockIdx.x * blockDim.x + threadIdx.x;
    if (i < n) deg[i] = fmaxf(deg[i], 1.0f);
}

// ---------------------------------------------------------------------------
// src_proj = x_src @ W_src using full-precision V_WMMA_F32_16X16X4_F32.
//   block = 256 threads = 8 waves; block b computes rows [16b, 16b+16);
//   wave w computes cols [16w, 16w+16). 64 WMMA steps over K=256.
// VGPR layouts (ISA 7.12.2, 32-bit):
//   A 16x4 : lane half h=lane/16, m=lane%16;  a[v] = A[m][k + v + 2h]
//   B 4x16 : n=lane%16;                        b[v] = W[k + v + 2h][n]
//   C 16x16: c[v] -> row (v + 8h), col n
// EXEC is all-1s here (no branches), as WMMA requires.
// ---------------------------------------------------------------------------
__global__ __launch_bounds__(256) void gemm_src_proj(
    const float* __restrict__ A,   // [N_SRC, D_SRC]
    const float* __restrict__ W,   // [D_SRC, D_REL]
    float* __restrict__ C)         // [N_SRC, D_REL]
{
    const int lane = threadIdx.x & 31;
    const int wave = threadIdx.x >> 5;     // 0..7 -> column tile
    const int m0   = blockIdx.x * 16;
    const int n0   = wave * 16;
    const int hi   = lane >> 4;            // 0 or 1
    const int l16  = lane & 15;

    const float* arow = A + (size_t)(m0 + l16) * D_SRC;  // row of A for this lane
    const float* wcol = W + n0 + l16;                    // column of W for this lane

    v8f c = {};
    #pragma unroll
    for (int k = 0; k < D_SRC; k += 4) {
        const int kk = k + 2 * hi;
        v2f a, b;
        a[0] = arow[kk + 0];
        a[1] = arow[kk + 1];
        b[0] = wcol[(size_t)(kk + 0) * D_REL];
        b[1] = wcol[(size_t)(kk + 1) * D_REL];
        c = __builtin_amdgcn_wmma_f32_16x16x4_f32(
                /*neg_a=*/false, a, /*neg_b=*/false, b,
                /*c_mod=*/(short)0, c, /*reuse_a=*/false, /*reuse_b=*/false);
    }

    float* cbase = C + n0 + l16;
    #pragma unroll
    for (int v = 0; v < 8; ++v) {
        cbase[(size_t)(m0 + v + 8 * hi) * D_REL] = c[v];
    }
}

// ---------------------------------------------------------------------------
// Mean-aggregation scatter: one wave per edge. Each lane gathers a float4 of
// src_proj[col] (L2-resident: 51 MB << 192 MB L2) and issues 4
// global_atomic_add_f32 into dst[row].
// ---------------------------------------------------------------------------
__global__ __launch_bounds__(256) void scatter_mean(
    const int* __restrict__ row, const int* __restrict__ col,
    const float* __restrict__ src_proj, const float* __restrict__ deg,
    float* __restrict__ dst, int e)
{
    const int wid  = (int)((blockIdx.x * blockDim.x + threadIdx.x) >> 5);
    const int lane = threadIdx.x & 31;
    if (wid >= e) return;

    const int r = row[wid];
    const int s = col[wid];
    const float scale = 1.0f / deg[r];     // deg already clamped >= 1

    const float4 v = ((const float4*)(src_proj + (size_t)s * D_REL))[lane];
    float* d = dst + (size_t)r * D_REL + lane * 4;
    atomicAdd(d + 0, v.x * scale);
    atomicAdd(d + 1, v.y * scale);
    atomicAdd(d + 2, v.z * scale);
    atomicAdd(d + 3, v.w * scale);
}

// ---------------------------------------------------------------------------
// d_in order (setup_inputs): x_src, x_dst, W_src, W_dst, edge_row, edge_col
// d_out: [dst (N_DST*D_REL) | src_proj (N_SRC*D_REL) | deg (N_DST)]
// ---------------------------------------------------------------------------
extern "C" void kernel_launch(void* const* d_in, const int* in_sizes, int n_in,
                              void* d_out, int out_size, void* d_ws, size_t ws_size,
                              hipStream_t stream) {
    const float* x_src = (const float*)d_in[0];
    const float* W_src = (const float*)d_in[2];
    const int*   erow  = (const int*)d_in[4];
    const int*   ecol  = (const int*)d_in[5];

    float* out      = (float*)d_out;
    float* dst      = out;                                   // N_DST * D_REL
    float* src_proj = out + (size_t)N_DST * D_REL;           // N_SRC * D_REL
    float* deg      = out + (size_t)N_DST * D_REL + (size_t)N_SRC * D_REL;  // N_DST

    const int dst_elems = N_DST * D_REL;

    // 1) zero accumulation targets
    zero_f32<<<(dst_elems + 255) / 256, 256, 0, stream>>>(dst, dst_elems);
    zero_f32<<<(N_DST + 255) / 256, 256, 0, stream>>>(deg, N_DST);

    // 2) degree count + clamp (deg section becomes the final deg output)
    count_deg<<<(NEDGE + 255) / 256, 256, 0, stream>>>(erow, deg, NEDGE);
    clamp_deg<<<(N_DST + 255) / 256, 256, 0, stream>>>(deg, N_DST);

    // 3) WMMA GEMM: 6250 blocks x (16 rows, 8 waves x 16 cols)
    gemm_src_proj<<<N_SRC / 16, 256, 0, stream>>>(x_src, W_src, src_proj);

    // 4) scatter-mean: one wave per edge -> 640000/8 blocks of 256
    scatter_mean<<<NEDGE / 8, 256, 0, stream>>>(erow, ecol, src_proj, deg, dst, NEDGE);
}